// TopKAttention_2293512536372
// MI455X (gfx1250) — compile-verified
//
#include <hip/hip_runtime.h>
#include <hip/hip_bf16.h>
#include <math.h>

// ---------------------------------------------------------------------------
// TopKAttention for MI455X (gfx1250), fp32 via V_WMMA_F32_16X16X4_F32,
// async global->LDS staging (ASYNCcnt) when the toolchain exposes it,
// ping-pong double-buffered LDS tiles, split-K for the W@v GEMM.
// B=16, N=2048, D=768, K=128.
// ---------------------------------------------------------------------------

typedef float v2f __attribute__((ext_vector_type(2)));
typedef float v8f __attribute__((ext_vector_type(8)));
typedef int   v4i __attribute__((ext_vector_type(4)));

#define BN    16
#define NN    2048
#define DD    768
#define KK    128
#define ROWS  (BN * NN)          // 32768
#define LDA   20                 // padded LDS stride for A tiles (conflict-free, 16B aligned)
#define LDB   128                // LDS stride for B tiles
#define ZSPLIT 8                 // split-K factor for Z = W @ v

// ---- gfx1250 async global->LDS (B128), gated so compile never breaks -------
#if defined(__has_builtin)
#if __has_builtin(__builtin_amdgcn_global_load_async_to_lds_b128)
#define HAVE_ASYNC 1
#endif
#endif
#ifndef HAVE_ASYNC
#define HAVE_ASYNC 0
#endif

typedef __attribute__((address_space(1))) v4i gv4i;
typedef __attribute__((address_space(3))) v4i lv4i;

__device__ __forceinline__ void cp16(float* lds_dst, const float* gsrc) {
#if HAVE_ASYNC
    __builtin_amdgcn_global_load_async_to_lds_b128(
        (gv4i*)gsrc, (lv4i*)lds_dst, /*offset=*/0, /*cpol=*/0);
#else
    *(float4*)lds_dst = *(const float4*)gsrc;
#endif
}

__device__ __forceinline__ void async_wait0() {
#if HAVE_ASYNC
#if defined(__has_builtin) && __has_builtin(__builtin_amdgcn_s_wait_asynccnt)
    __builtin_amdgcn_s_wait_asynccnt(0);
#else
    asm volatile("s_wait_asynccnt 0x0" ::: "memory");
#endif
#endif
}

// D = A(16x4) * B(4x16) + C, fp32 WMMA. One 16x16 subtile per call.
__device__ __forceinline__ v8f wmma_f32(v2f a, v2f b, v8f c) {
    return __builtin_amdgcn_wmma_f32_16x16x4_f32(
        /*neg_a=*/false, a, /*neg_b=*/false, b,
        /*c_mod=*/(short)0, c, /*reuse_a=*/false, /*reuse_b=*/false);
}

// Per-wave 64x32 patch of the 128x128 tile: 4 (M) x 2 (N) subtiles of 16x16.
// K-slab of 16 consumed as 4 chained K=4 WMMAs.
#define TILE_COMPUTE(As, Bs)                                                   \
    do {                                                                       \
        const int rowb_ = lane & 15;                                           \
        const int klo_  = (lane >> 4) << 1;                                    \
        _Pragma("unroll")                                                      \
        for (int kk_ = 0; kk_ < 16; kk_ += 4) {                                \
            v2f afr_[4];                                                       \
            v2f bfr_[2];                                                       \
            _Pragma("unroll")                                                  \
            for (int mi_ = 0; mi_ < 4; ++mi_) {                                \
                const int r_ = waveM + mi_ * 16 + rowb_;                       \
                afr_[mi_][0] = (As)[r_ * LDA + kk_ + klo_];                    \
                afr_[mi_][1] = (As)[r_ * LDA + kk_ + klo_ + 1];                \
            }                                                                  \
            _Pragma("unroll")                                                  \
            for (int ni_ = 0; ni_ < 2; ++ni_) {                                \
                const int c_ = waveN + ni_ * 16 + rowb_;                       \
                bfr_[ni_][0] = (Bs)[(kk_ + klo_) * LDB + c_];                  \
                bfr_[ni_][1] = (Bs)[(kk_ + klo_ + 1) * LDB + c_];              \
            }                                                                  \
            _Pragma("unroll")                                                  \
            for (int mi_ = 0; mi_ < 4; ++mi_)                                  \
                _Pragma("unroll")                                              \
                for (int ni_ = 0; ni_ < 2; ++ni_)                              \
                    acc[mi_][ni_] = wmma_f32(afr_[mi_], bfr_[ni_], acc[mi_][ni_]); \
        }                                                                      \
    } while (0)

#define WAVE_SETUP()                                                           \
    const int tid   = threadIdx.x;                                             \
    const int lane  = tid & 31;                                                \
    const int w     = tid >> 5;                                                \
    const int waveM = (w >> 2) * 64;                                           \
    const int waveN = (w & 3) * 32

// ---------------------------------------------------------------------------
// Per-row LayerNorm stats: stats[2r] = mu, stats[2r+1] = rsqrt(var + 1e-5).
// If addB != 0, operates on (A + Badd). One block per row.
// ---------------------------------------------------------------------------
__global__ __launch_bounds__(256) void ln_stats_kernel(
    const float* __restrict__ A, const float* __restrict__ Badd,
    const int addB, float* __restrict__ stats) {
    const int row = blockIdx.x;
    const int tid = threadIdx.x;
    __shared__ float r1[256];
    __shared__ float r2[256];
    float s = 0.f, sq = 0.f;
    for (int c = tid; c < DD; c += 256) {
        float v = A[(size_t)row * DD + c];
        if (addB) v += Badd[(size_t)row * DD + c];
        s += v;
        sq += v * v;
    }
    r1[tid] = s;
    r2[tid] = sq;
    __syncthreads();
    for (int st = 128; st > 0; st >>= 1) {
        if (tid < st) { r1[tid] += r1[tid + st]; r2[tid] += r2[tid + st]; }
        __syncthreads();
    }
    if (tid == 0) {
        const float mu  = r1[0] * (1.0f / DD);
        const float var = r2[0] * (1.0f / DD) - mu * mu;
        stats[2 * row]     = mu;
        stats[2 * row + 1] = rsqrtf(var + 1e-5f);
    }
}

// ---------------------------------------------------------------------------
// GEMM 1: qkv = LN(x) @ Wqkv + bqkv, LN fused into A-tile load.
// A tiles: VALU-fused LN -> ds_store; B tiles: async global->LDS.
// Ping-pong double buffering. grid = (18 col tiles, 256 row tiles).
// ---------------------------------------------------------------------------
__global__ __launch_bounds__(256) void gemm_qkv_kernel(
    const float* __restrict__ x, const float* __restrict__ stats,
    const float* __restrict__ g, const float* __restrict__ be,
    const float* __restrict__ Wqkv, const float* __restrict__ bqkv,
    float* __restrict__ qb, float* __restrict__ kb, float* __restrict__ vb) {
    __shared__ __align__(16) float As[2][128 * LDA];
    __shared__ __align__(16) float Bs[2][16 * LDB];
    WAVE_SETUP();
    const int cTile = blockIdx.x * 128;
    const int rTile = blockIdx.y * 128;
    v8f acc[4][2] = {};

    auto loadA = [&](int k0, int buf) {
        for (int f = tid; f < 512; f += 256) {
            const int row = f >> 2;
            const int c4  = (f & 3) * 4;
            const int r   = rTile + row;
            const float4 xv = *(const float4*)(x + (size_t)r * DD + k0 + c4);
            const float4 gv = *(const float4*)(g + k0 + c4);
            const float4 bv = *(const float4*)(be + k0 + c4);
            const float mu = stats[2 * r];
            const float rs = stats[2 * r + 1];
            float4 hv;
            hv.x = (xv.x - mu) * rs * gv.x + bv.x;
            hv.y = (xv.y - mu) * rs * gv.y + bv.y;
            hv.z = (xv.z - mu) * rs * gv.z + bv.z;
            hv.w = (xv.w - mu) * rs * gv.w + bv.w;
            *(float4*)(&As[buf][row * LDA + c4]) = hv;
        }
    };
    auto loadB = [&](int k0, int buf) {
        for (int f = tid; f < 512; f += 256) {
            const int row = f >> 5;
            const int c4  = (f & 31) * 4;
            cp16(&Bs[buf][row * LDB + c4],
                 Wqkv + (size_t)(k0 + row) * (3 * DD) + cTile + c4);
        }
    };

    loadA(0, 0);
    loadB(0, 0);
    async_wait0();
    __syncthreads();
    int cur = 0;
    for (int k0 = 0; k0 < DD; k0 += 16) {
        const int nxt = k0 + 16;
        if (nxt < DD) { loadA(nxt, cur ^ 1); loadB(nxt, cur ^ 1); }
        TILE_COMPUTE(&As[cur][0], &Bs[cur][0]);
        async_wait0();
        __syncthreads();
        cur ^= 1;
    }

    float* outp;
    int ccol;
    if (cTile < DD)          { outp = qb; ccol = cTile; }
    else if (cTile < 2 * DD) { outp = kb; ccol = cTile - DD; }
    else                     { outp = vb; ccol = cTile - 2 * DD; }
    const int rowb = lane & 15;
    const int rhi  = (lane >> 4) * 8;
#pragma unroll
    for (int mi = 0; mi < 4; ++mi)
#pragma unroll
        for (int ni = 0; ni < 2; ++ni)
#pragma unroll
            for (int i = 0; i < 8; ++i) {
                const int r = rTile + waveM + mi * 16 + rhi + i;
                const int c = waveN + ni * 16 + rowb;
                outp[(size_t)r * DD + ccol + c] = acc[mi][ni][i] + bqkv[cTile + c];
            }
}

// ---------------------------------------------------------------------------
// s[r] = scale * sum_d q[r,d]*k[r,d]. One block per row.
// ---------------------------------------------------------------------------
__global__ __launch_bounds__(256) void dot_s_kernel(
    const float* __restrict__ q, const float* __restrict__ k,
    const float* __restrict__ scale_p, float* __restrict__ s) {
    const int row = blockIdx.x;
    const int tid = threadIdx.x;
    __shared__ float red[256];
    float p = 0.f;
    for (int c = tid; c < DD; c += 256)
        p += q[(size_t)row * DD + c] * k[(size_t)row * DD + c];
    red[tid] = p;
    __syncthreads();
    for (int st = 128; st > 0; st >>= 1) {
        if (tid < st) red[tid] += red[tid + st];
        __syncthreads();
    }
    if (tid == 0) s[row] = red[0] * scale_p[0];
}

// ---------------------------------------------------------------------------
// Iterative relaxed top-k: K=128 sequential softmax rounds over N=2048.
// One block per batch; logits in registers; wave32 shuffle butterflies +
// 16-entry LDS cross-wave stage (4 barriers per round).
// ---------------------------------------------------------------------------
__device__ __forceinline__ float wred_max(float v) {
#pragma unroll
    for (int m = 16; m > 0; m >>= 1) v = fmaxf(v, __shfl_xor(v, m, 32));
    return v;
}
__device__ __forceinline__ float wred_sum(float v) {
#pragma unroll
    for (int m = 16; m > 0; m >>= 1) v += __shfl_xor(v, m, 32);
    return v;
}

__global__ __launch_bounds__(512) void topk_kernel(
    const float* __restrict__ s, const float* __restrict__ tau_p,
    float* __restrict__ Wout) {
    const int b    = blockIdx.x;
    const int tid  = threadIdx.x;
    const int lane = tid & 31;
    const int wid  = tid >> 5;     // 0..15
    __shared__ float redm[16];
    __shared__ float reds[16];
    float l[4];
#pragma unroll
    for (int t = 0; t < 4; ++t) l[t] = s[(size_t)b * NN + tid + 512 * t];
    float tp = tau_p[0];
    tp = fminf(fmaxf(tp, -1.4f), 5.0f);
    const float tau    = 1.0f / (1.0f + expf(-tp));
    const float invtau = 1.0f / tau;

    for (int k = 0; k < KK; ++k) {
        // ---- block max ----
        float m = fmaxf(fmaxf(l[0], l[1]), fmaxf(l[2], l[3]));
        m = wred_max(m);
        if (lane == 0) redm[wid] = m;
        __syncthreads();
        if (wid == 0) {
            float t = (lane < 16) ? redm[lane] : -3.402823466e38f;
            t = wred_max(t);
            if (lane == 0) redm[0] = t;
        }
        __syncthreads();
        const float gmax = redm[0];
        // ---- exp + block sum ----
        float e[4];
        float ls = 0.f;
#pragma unroll
        for (int t = 0; t < 4; ++t) { e[t] = expf((l[t] - gmax) * invtau); ls += e[t]; }
        ls = wred_sum(ls);
        if (lane == 0) reds[wid] = ls;
        __syncthreads();
        if (wid == 0) {
            float t = (lane < 16) ? reds[lane] : 0.f;
            t = wred_sum(t);
            if (lane == 0) reds[0] = t;
        }
        __syncthreads();
        const float inv = 1.0f / reds[0];
#pragma unroll
        for (int t = 0; t < 4; ++t) {
            const float wv = e[t] * inv;
            Wout[((size_t)b * KK + k) * NN + tid + 512 * t] = wv;
            l[t] += logf(1.0f - wv + 1e-10f);
        }
    }
}

// ---------------------------------------------------------------------------
// GEMM 2 (batched, split-K): Zp[sk][b] = W[b][:, skK] @ v[b][skK, :].
// Both tiles are plain copies -> async global->LDS. grid = (6, ZSPLIT, 16).
// ---------------------------------------------------------------------------
__global__ __launch_bounds__(256) void gemm_z_kernel(
    const float* __restrict__ Wb, const float* __restrict__ vb,
    float* __restrict__ Zp) {
    __shared__ __align__(16) float As[2][128 * LDA];
    __shared__ __align__(16) float Bs[2][16 * LDB];
    WAVE_SETUP();
    const int cTile = blockIdx.x * 128;
    const int sk    = blockIdx.y;
    const int b     = blockIdx.z;
    const int kBeg  = sk * (NN / ZSPLIT);
    const int kEnd  = kBeg + (NN / ZSPLIT);
    const float* A  = Wb + (size_t)b * KK * NN;
    const float* Bm = vb + (size_t)b * NN * DD;
    v8f acc[4][2] = {};

    auto loadA = [&](int k0, int buf) {
        for (int f = tid; f < 512; f += 256) {
            const int row = f >> 2;
            const int c4  = (f & 3) * 4;
            cp16(&As[buf][row * LDA + c4], A + (size_t)row * NN + k0 + c4);
        }
    };
    auto loadB = [&](int k0, int buf) {
        for (int f = tid; f < 512; f += 256) {
            const int row = f >> 5;
            const int c4  = (f & 31) * 4;
            cp16(&Bs[buf][row * LDB + c4], Bm + (size_t)(k0 + row) * DD + cTile + c4);
        }
    };

    loadA(kBeg, 0);
    loadB(kBeg, 0);
    async_wait0();
    __syncthreads();
    int cur = 0;
    for (int k0 = kBeg; k0 < kEnd; k0 += 16) {
        const int nxt = k0 + 16;
        if (nxt < kEnd) { loadA(nxt, cur ^ 1); loadB(nxt, cur ^ 1); }
        TILE_COMPUTE(&As[cur][0], &Bs[cur][0]);
        async_wait0();
        __syncthreads();
        cur ^= 1;
    }

    const int rowb = lane & 15;
    const int rhi  = (lane >> 4) * 8;
    float* Zout = Zp + (size_t)sk * BN * KK * DD + (size_t)b * KK * DD;
#pragma unroll
    for (int mi = 0; mi < 4; ++mi)
#pragma unroll
        for (int ni = 0; ni < 2; ++ni)
#pragma unroll
            for (int i = 0; i < 8; ++i) {
                const int r = waveM + mi * 16 + rhi + i;
                const int c = cTile + waveN + ni * 16 + rowb;
                Zout[(size_t)r * DD + c] = acc[mi][ni][i];
            }
}

// Deterministic split-K reduction: Z = sum over ZSPLIT partials.
__global__ __launch_bounds__(256) void zred_kernel(
    const float* __restrict__ Zp, float* __restrict__ Z) {
    const size_t i = (size_t)blockIdx.x * 256 + threadIdx.x;
    const size_t STRIDE = (size_t)BN * KK * DD;
    float acc = 0.f;
#pragma unroll
    for (int p = 0; p < ZSPLIT; ++p) acc += Zp[p * STRIDE + i];
    Z[i] = acc;
}

// ---------------------------------------------------------------------------
// GEMM 3 (batched): Y[b] = W[b]^T(2048x128) @ Z[b](128x768); A loaded
// transposed (coalesced along m, scatter to LDS); B async.
// grid = (6, 16 row tiles, 16 batches).
// ---------------------------------------------------------------------------
__global__ __launch_bounds__(256) void gemm_y_kernel(
    const float* __restrict__ Wb, const float* __restrict__ Z,
    float* __restrict__ Y) {
    __shared__ __align__(16) float As[2][128 * LDA];
    __shared__ __align__(16) float Bs[2][16 * LDB];
    WAVE_SETUP();
    const int cTile = blockIdx.x * 128;
    const int rTile = blockIdx.y * 128;
    const int b     = blockIdx.z;
    const float* A  = Wb + (size_t)b * KK * NN;   // [128, 2048]
    const float* Bm = Z + (size_t)b * KK * DD;    // [128, 768]
    v8f acc[4][2] = {};

    auto loadA = [&](int k0, int buf) {           // transposed scatter
        for (int f = tid; f < 512; f += 256) {
            const int kd = f >> 5;
            const int m4 = (f & 31) * 4;
            const float4 wv = *(const float4*)(A + (size_t)(k0 + kd) * NN + rTile + m4);
            As[buf][(m4 + 0) * LDA + kd] = wv.x;
            As[buf][(m4 + 1) * LDA + kd] = wv.y;
            As[buf][(m4 + 2) * LDA + kd] = wv.z;
            As[buf][(m4 + 3) * LDA + kd] = wv.w;
        }
    };
    auto loadB = [&](int k0, int buf) {
        for (int f = tid; f < 512; f += 256) {
            const int row = f >> 5;
            const int c4  = (f & 31) * 4;
            cp16(&Bs[buf][row * LDB + c4], Bm + (size_t)(k0 + row) * DD + cTile + c4);
        }
    };

    loadA(0, 0);
    loadB(0, 0);
    async_wait0();
    __syncthreads();
    int cur = 0;
    for (int k0 = 0; k0 < KK; k0 += 16) {
        const int nxt = k0 + 16;
        if (nxt < KK) { loadA(nxt, cur ^ 1); loadB(nxt, cur ^ 1); }
        TILE_COMPUTE(&As[cur][0], &Bs[cur][0]);
        async_wait0();
        __syncthreads();
        cur ^= 1;
    }

    const int rowb = lane & 15;
    const int rhi  = (lane >> 4) * 8;
#pragma unroll
    for (int mi = 0; mi < 4; ++mi)
#pragma unroll
        for (int ni = 0; ni < 2; ++ni)
#pragma unroll
            for (int i = 0; i < 8; ++i) {
                const int r = b * NN + rTile + waveM + mi * 16 + rhi + i;
                const int c = cTile + waveN + ni * 16 + rowb;
                Y[(size_t)r * DD + c] = acc[mi][ni][i];
            }
}

// ---------------------------------------------------------------------------
// GEMM 4: out = gelu( LN(Y+v) @ Wo + bo ), LN fused into A load, exact GELU.
// grid = (6 col tiles, 256 row tiles).
// ---------------------------------------------------------------------------
__global__ __launch_bounds__(256) void gemm_out_kernel(
    const float* __restrict__ Y, const float* __restrict__ vb,
    const float* __restrict__ stats, const float* __restrict__ g,
    const float* __restrict__ be, const float* __restrict__ Wo,
    const float* __restrict__ bo, float* __restrict__ out) {
    __shared__ __align__(16) float As[2][128 * LDA];
    __shared__ __align__(16) float Bs[2][16 * LDB];
    WAVE_SETUP();
    const int cTile = blockIdx.x * 128;
    const int rTile = blockIdx.y * 128;
    v8f acc[4][2] = {};

    auto loadA = [&](int k0, int buf) {
        for (int f = tid; f < 512; f += 256) {
            const int row = f >> 2;
            const int c4  = (f & 3) * 4;
            const int r   = rTile + row;
            const float4 yv = *(const float4*)(Y + (size_t)r * DD + k0 + c4);
            const float4 vv = *(const float4*)(vb + (size_t)r * DD + k0 + c4);
            const float4 gv = *(const float4*)(g + k0 + c4);
            const float4 bv = *(const float4*)(be + k0 + c4);
            const float mu = stats[2 * r];
            const float rs = stats[2 * r + 1];
            float4 hv;
            hv.x = ((yv.x + vv.x) - mu) * rs * gv.x + bv.x;
            hv.y = ((yv.y + vv.y) - mu) * rs * gv.y + bv.y;
            hv.z = ((yv.z + vv.z) - mu) * rs * gv.z + bv.z;
            hv.w = ((yv.w + vv.w) - mu) * rs * gv.w + bv.w;
            *(float4*)(&As[buf][row * LDA + c4]) = hv;
        }
    };
    auto loadB = [&](int k0, int buf) {
        for (int f = tid; f < 512; f += 256) {
            const int row = f >> 5;
            const int c4  = (f & 31) * 4;
            cp16(&Bs[buf][row * LDB + c4], Wo + (size_t)(k0 + row) * DD + cTile + c4);
        }
    };

    loadA(0, 0);
    loadB(0, 0);
    async_wait0();
    __syncthreads();
    int cur = 0;
    for (int k0 = 0; k0 < DD; k0 += 16) {
        const int nxt = k0 + 16;
        if (nxt < DD) { loadA(nxt, cur ^ 1); loadB(nxt, cur ^ 1); }
        TILE_COMPUTE(&As[cur][0], &Bs[cur][0]);
        async_wait0();
        __syncthreads();
        cur ^= 1;
    }

    const int rowb = lane & 15;
    const int rhi  = (lane >> 4) * 8;
#pragma unroll
    for (int mi = 0; mi < 4; ++mi)
#pragma unroll
        for (int ni = 0; ni < 2; ++ni)
#pragma unroll
            for (int i = 0; i < 8; ++i) {
                const int r = rTile + waveM + mi * 16 + rhi + i;
                const int c = cTile + waveN + ni * 16 + rowb;
                float val = acc[mi][ni][i] + bo[c];
                val = 0.5f * val * (1.0f + erff(val * 0.70710678118654752f));
                out[(size_t)r * DD + c] = val;
            }
}

// ---------------------------------------------------------------------------
// Host-side orchestration (graph-capture safe: only kernel launches on stream)
// ---------------------------------------------------------------------------
extern "C" void kernel_launch(void* const* d_in, const int* in_sizes, int n_in,
                              void* d_out, int out_size, void* d_ws, size_t ws_size,
                              hipStream_t stream) {
    const float* x       = (const float*)d_in[0];
    const float* ln1_g   = (const float*)d_in[1];
    const float* ln1_b   = (const float*)d_in[2];
    const float* Wqkv    = (const float*)d_in[3];
    const float* bqkv    = (const float*)d_in[4];
    const float* lno_g   = (const float*)d_in[5];
    const float* lno_b   = (const float*)d_in[6];
    const float* Wo      = (const float*)d_in[7];
    const float* bo      = (const float*)d_in[8];
    const float* tau_p   = (const float*)d_in[9];
    const float* scale_p = (const float*)d_in[10];

    float* ws = (float*)d_ws;
    const size_t BND = (size_t)ROWS * DD;      // 25,165,824 floats
    float* qb  = ws;                           // [32768, 768]  (reused as Y)
    float* kb  = ws + BND;                     // [32768, 768]
    float* vb  = ws + 2 * BND;                 // [32768, 768]
    float* sb  = ws + 3 * BND;                 // [32768]
    float* Wt  = sb + ROWS;                    // [16, 128, 2048]
    float* Zb  = Wt + (size_t)BN * KK * NN;    // [16, 128, 768]
    float* st1 = Zb + (size_t)BN * KK * DD;    // [32768, 2]
    float* st2 = st1 + 2 * (size_t)ROWS;       // [32768, 2]
    float* Zp  = st2 + 2 * (size_t)ROWS;       // [ZSPLIT, 16, 128, 768]
    float* Yb  = qb;                           // alias: q dead after dot_s

    // 1) LN1 row stats over x
    ln_stats_kernel<<<ROWS, 256, 0, stream>>>(x, nullptr, 0, st1);
    // 2) qkv = LN(x) @ Wqkv + bqkv  (fp32 WMMA, fused LN, async B tiles)
    gemm_qkv_kernel<<<dim3(18, 256), 256, 0, stream>>>(x, st1, ln1_g, ln1_b,
                                                       Wqkv, bqkv, qb, kb, vb);
    // 3) s = scale * sum(q*k)
    dot_s_kernel<<<ROWS, 256, 0, stream>>>(qb, kb, scale_p, sb);
    // 4) W = diff_topk(s, tau)  (sequential K rounds, one WG per batch)
    topk_kernel<<<BN, 512, 0, stream>>>(sb, tau_p, Wt);
    // 5) Zp[sk] = W[:, skK] @ v[skK, :]  (split-K, async tiles)
    gemm_z_kernel<<<dim3(6, ZSPLIT, BN), 256, 0, stream>>>(Wt, vb, Zp);
    //    Z = sum(Zp)
    zred_kernel<<<(BN * KK * DD) / 256, 256, 0, stream>>>(Zp, Zb);
    // 6) Y = W^T @ Z   (writes over q region)
    gemm_y_kernel<<<dim3(6, 16, BN), 256, 0, stream>>>(Wt, Zb, Yb);
    // 7) LN stats over (Y + v)
    ln_stats_kernel<<<ROWS, 256, 0, stream>>>(Yb, vb, 1, st2);
    // 8) out = gelu( LN(Y+v) @ Wo + bo )
    gemm_out_kernel<<<dim3(6, 256), 256, 0, stream>>>(Yb, vb, st2, lno_g, lno_b,
                                                      Wo, bo, (float*)d_out);
}